// MPARBlock_77781857730612
// MI455X (gfx1250) — compile-verified
//
#include <hip/hip_runtime.h>
#include <math.h>

// MPAR block for MI455X (gfx1250, wave32).
// B=16, C=128, H=W=128. All fp32.
//
// Pipeline:
//   k_cw       : cw[w] = sum_k Dw[k,w]  (DCT column sums; freq branch collapses)
//   k_reduce   : per (b,c): gap = mean(x), fG = mean(x * cw[w]*cw[h])
//   k_mask     : m[b,p] = sum_c x*gc_mask_w + gc_mask_b
//   k_softmax  : softmax over HW, in place
//   k_ctx      : ctx[b,c] = sum_p x * m
//   k_mlp      : ca/fwt/gc MLPs -> s[b,c], bias2[b,o], wts=softmax(alpha)
//   k_conv     : g[b,p] = sigmoid(9x9 conv + b)    (LDS-tiled direct conv)
//   k_gemm     : out = (W.s) x  +  w1*g*(W x)  +  bias2   via V_WMMA_F32_16X16X4_F32

typedef float v2f __attribute__((ext_vector_type(2)));
typedef float v8f __attribute__((ext_vector_type(8)));

#define BB 16
#define CC 128
#define HW 16384           // 128*128
#define PIf 3.14159265358979323846f

// ---------------- workspace layout (floats) ----------------
#define CW_OFF     0          // 128
#define GAP_OFF    128        // 2048
#define FG_OFF     2176       // 2048
#define CTX_OFF    4224       // 2048
#define S_OFF      6272       // 2048
#define BIAS2_OFF  8320       // 2048
#define WTS_OFF    10368      // 4
#define M_OFF      16384      // 262144
#define G_OFF      278528     // 262144

// ---------------------------------------------------------------------------
__global__ void k_cw(float* __restrict__ cw) {
  int w = threadIdx.x;                       // 0..127
  float s = 0.f;
  for (int k = 0; k < 128; ++k)
    s += 2.f * cosf(PIf * (2.f * w + 1.f) * (float)k * (1.f / 256.f));
  cw[w] = s;
}

// per (b,c): gap and weighted-freq gap, one pass over the plane
__global__ void __launch_bounds__(256) k_reduce(
    const float* __restrict__ x, const float* __restrict__ cw,
    float* __restrict__ gap, float* __restrict__ fG) {
  __shared__ float cwl[128];
  __shared__ float sg[256], sf[256];
  int bc = blockIdx.x, tid = threadIdx.x;
  if (tid < 128) cwl[tid] = cw[tid];
  __syncthreads();
  const float* xp = x + ((size_t)bc << 14);
  float ag = 0.f, af = 0.f;
  for (int i = 0; i < 64; ++i) {
    int p = tid + (i << 8);
    float v = xp[p];
    ag += v;
    af += v * cwl[p & 127] * cwl[p >> 7];
  }
  sg[tid] = ag; sf[tid] = af;
  __syncthreads();
  for (int s = 128; s > 0; s >>= 1) {
    if (tid < s) { sg[tid] += sg[tid + s]; sf[tid] += sf[tid + s]; }
    __syncthreads();
  }
  if (tid == 0) {
    gap[bc] = sg[0] * (1.f / 16384.f);
    fG[bc]  = sf[0] * (1.f / 16384.f);
  }
}

// mask logits: m[b,p] = sum_c x[b,c,p]*gw[c] + gb
__global__ void __launch_bounds__(256) k_mask(
    const float* __restrict__ x, const float* __restrict__ gw,
    const float* __restrict__ gb, float* __restrict__ m) {
  int blk = blockIdx.x;
  int b = blk >> 6;
  int p = ((blk & 63) << 8) + threadIdx.x;
  const float* xp = x + ((size_t)b << 21) + p;
  float acc = 0.f;
  for (int c = 0; c < 128; ++c) acc += xp[(size_t)c << 14] * gw[c];
  m[(b << 14) + p] = acc + gb[0];
}

__global__ void __launch_bounds__(256) k_softmax(float* __restrict__ m) {
  __shared__ float red[256];
  int b = blockIdx.x, tid = threadIdx.x;
  float* mp = m + (b << 14);
  float mx = -1e30f;
  for (int i = 0; i < 64; ++i) mx = fmaxf(mx, mp[tid + (i << 8)]);
  red[tid] = mx; __syncthreads();
  for (int s = 128; s > 0; s >>= 1) {
    if (tid < s) red[tid] = fmaxf(red[tid], red[tid + s]);
    __syncthreads();
  }
  mx = red[0]; __syncthreads();
  float sum = 0.f;
  for (int i = 0; i < 64; ++i) sum += __expf(mp[tid + (i << 8)] - mx);
  red[tid] = sum; __syncthreads();
  for (int s = 128; s > 0; s >>= 1) {
    if (tid < s) red[tid] += red[tid + s];
    __syncthreads();
  }
  float inv = 1.f / red[0];
  for (int i = 0; i < 64; ++i) {
    int p = tid + (i << 8);
    mp[p] = __expf(mp[p] - mx) * inv;
  }
}

// ctx[b,c] = sum_p x * softmax_mask
__global__ void __launch_bounds__(256) k_ctx(
    const float* __restrict__ x, const float* __restrict__ m,
    float* __restrict__ ctx) {
  __shared__ float red[256];
  int bc = blockIdx.x, tid = threadIdx.x;
  int b = bc >> 7;
  const float* xp = x + ((size_t)bc << 14);
  const float* mp = m + (b << 14);
  float acc = 0.f;
  for (int i = 0; i < 64; ++i) { int p = tid + (i << 8); acc += xp[p] * mp[p]; }
  red[tid] = acc; __syncthreads();
  for (int s = 128; s > 0; s >>= 1) {
    if (tid < s) red[tid] += red[tid + s];
    __syncthreads();
  }
  if (tid == 0) ctx[bc] = red[0];
}

// all the tiny [B,C] MLP math; one block per b, 128 threads (thread == channel)
__global__ void __launch_bounds__(128) k_mlp(
    const float* __restrict__ gap, const float* __restrict__ fG,
    const float* __restrict__ ctx,
    const float* __restrict__ ca_w1, const float* __restrict__ ca_b1,
    const float* __restrict__ ca_w2, const float* __restrict__ ca_b2,
    const float* __restrict__ fa_w1, const float* __restrict__ fa_b1,
    const float* __restrict__ fa_w2, const float* __restrict__ fa_b2,
    const float* __restrict__ t1w, const float* __restrict__ t1b,
    const float* __restrict__ lng, const float* __restrict__ lnb,
    const float* __restrict__ t2w, const float* __restrict__ t2b,
    const float* __restrict__ alpha,
    const float* __restrict__ outw, const float* __restrict__ outb,
    float* __restrict__ sArr, float* __restrict__ bias2,
    float* __restrict__ wtsOut) {
  __shared__ float gl[128], fl[128], cl[128];
  __shared__ float h1[16], h2[16], tl[8], c2[128], wl[4], ms[2];
  int b = blockIdx.x, tid = threadIdx.x;
  gl[tid] = gap[(b << 7) + tid];
  fl[tid] = fG[(b << 7) + tid];
  cl[tid] = ctx[(b << 7) + tid];
  if (tid == 0) {
    float a0 = alpha[0], a1 = alpha[1], a2 = alpha[2], a3 = alpha[3];
    float mx = fmaxf(fmaxf(a0, a1), fmaxf(a2, a3));
    float e0 = __expf(a0 - mx), e1 = __expf(a1 - mx);
    float e2 = __expf(a2 - mx), e3 = __expf(a3 - mx);
    float inv = 1.f / (e0 + e1 + e2 + e3);
    wl[0] = e0 * inv; wl[1] = e1 * inv; wl[2] = e2 * inv; wl[3] = e3 * inv;
    if (b == 0) { wtsOut[0] = wl[0]; wtsOut[1] = wl[1]; wtsOut[2] = wl[2]; wtsOut[3] = wl[3]; }
  }
  __syncthreads();
  if (tid < 16) {
    float a = ca_b1[tid], f = fa_b1[tid];
    for (int c = 0; c < 128; ++c) {
      a += ca_w1[tid * 128 + c] * gl[c];
      f += fa_w1[tid * 128 + c] * fl[c];
    }
    h1[tid] = fmaxf(a, 0.f);
    h2[tid] = fmaxf(f, 0.f);
  }
  if (tid < 8) {
    float t = t1b[tid];
    for (int c = 0; c < 128; ++c) t += t1w[tid * 128 + c] * cl[c];
    tl[tid] = t;
  }
  __syncthreads();
  if (tid == 0) {
    float mu = 0.f;
    for (int j = 0; j < 8; ++j) mu += tl[j];
    mu *= 0.125f;
    float var = 0.f;
    for (int j = 0; j < 8; ++j) { float d = tl[j] - mu; var += d * d; }
    var *= 0.125f;
    ms[0] = mu; ms[1] = rsqrtf(var + 1e-5f);
  }
  __syncthreads();
  if (tid < 8) tl[tid] = fmaxf((tl[tid] - ms[0]) * ms[1] * lng[tid] + lnb[tid], 0.f);
  __syncthreads();
  {
    float ca = ca_b2[tid], fw = fa_b2[tid];
    for (int j = 0; j < 16; ++j) {
      ca += ca_w2[tid * 16 + j] * h1[j];
      fw += fa_w2[tid * 16 + j] * h2[j];
    }
    ca = 1.f / (1.f + __expf(-ca));
    fw = 1.f / (1.f + __expf(-fw));
    float c2v = t2b[tid];
    for (int j = 0; j < 8; ++j) c2v += t2w[tid * 8 + j] * tl[j];
    c2[tid] = c2v;
    sArr[(b << 7) + tid] = wl[0] * ca + wl[2] * fw + wl[3];
  }
  __syncthreads();
  {
    float acc = 0.f;
    for (int c = 0; c < 128; ++c) acc += outw[tid * 128 + c] * c2[c];
    bias2[(b << 7) + tid] = wl[3] * acc + outb[tid];
  }
}

// 9x9 spatial conv over all C -> sigmoid gate g[b,p]. 32x32 output tile/block.
__global__ void __launch_bounds__(256) k_conv(
    const float* __restrict__ x, const float* __restrict__ w,
    const float* __restrict__ wb, float* __restrict__ g) {
  __shared__ float wS[128 * 81];     // 41.5 KB: all conv weights resident
  __shared__ float tS[40 * 40];      // 32x32 tile + 4-halo
  int tid = threadIdx.x, blk = blockIdx.x;
  int b  = blk >> 4;
  int ty = ((blk >> 2) & 3) << 5;
  int tx = (blk & 3) << 5;
  for (int i = tid; i < 128 * 81; i += 256) wS[i] = w[i];
  float acc[4] = {0.f, 0.f, 0.f, 0.f};
  int ly[4], lx[4];
  for (int k = 0; k < 4; ++k) { int pix = tid + (k << 8); ly[k] = pix >> 5; lx[k] = pix & 31; }
  const float* xb = x + ((size_t)b << 21);
  for (int c = 0; c < 128; ++c) {
    __syncthreads();
    for (int i = tid; i < 1600; i += 256) {
      int yy = ty - 4 + i / 40;
      int xx = tx - 4 + i % 40;
      float v = 0.f;
      if (yy >= 0 && yy < 128 && xx >= 0 && xx < 128)
        v = xb[((size_t)c << 14) + (yy << 7) + xx];
      tS[i] = v;
    }
    __syncthreads();
    const float* wc = &wS[c * 81];
    for (int k = 0; k < 4; ++k) {
      float a = 0.f;
#pragma unroll
      for (int dy = 0; dy < 9; ++dy)
#pragma unroll
        for (int dx = 0; dx < 9; ++dx)
          a += tS[(ly[k] + dy) * 40 + lx[k] + dx] * wc[dy * 9 + dx];
      acc[k] += a;
    }
  }
  float bb = wb[0];
  for (int k = 0; k < 4; ++k) {
    float gv = 1.f / (1.f + __expf(-(acc[k] + bb)));
    g[(b << 14) + ((ty + ly[k]) << 7) + (tx + lx[k])] = gv;
  }
}

// Fused dual-GEMM epilogue kernel:
//   out[b,o,p] = [ (W .col-scaled-by s[b]) x ] + w1*g[b,p]*[ W x ] + bias2[b,o]
// per-block: full M=128, N=64 pixel tile, K=128 via V_WMMA_F32_16X16X4_F32.
#define XP 68  // padded LDS row stride (16B-aligned float4 rows, low-conflict)

__global__ void __launch_bounds__(256) k_gemm(
    const float* __restrict__ x, const float* __restrict__ A,
    const float* __restrict__ s, const float* __restrict__ bias2,
    const float* __restrict__ wts, const float* __restrict__ g,
    float* __restrict__ out) {
  __shared__ __align__(16) float Xs[128 * XP];
  __shared__ float sS[128];
  __shared__ float bS[128];
  int tid = threadIdx.x, blk = blockIdx.x;
  int b  = blk >> 8;            // 256 N-tiles per batch image
  int p0 = (blk & 255) << 6;    // 64-pixel tile
  const size_t xbase = ((size_t)b << 21) + p0;
  if (tid < 128) {
    sS[tid] = s[(b << 7) + tid];
    bS[tid] = bias2[(b << 7) + tid];
  }
  // stage x tile [K=128][N=64] via float4 (global_load_b128)
#pragma unroll
  for (int i = 0; i < 8; ++i) {
    int idx = (tid + (i << 8)) << 2;
    int c = idx >> 6, n = idx & 63;
    const float4 v = *(const float4*)(x + xbase + ((size_t)c << 14) + n);
    *(float4*)(&Xs[c * XP + n]) = v;
  }
  __syncthreads();
  int wave = tid >> 5;          // wave -> 16-row stripe of W (M subtile)
  int lane = tid & 31;
  int l15  = lane & 15;
  int hi   = lane >> 4;         // half-wave K split per ISA fragment layout
  int row  = (wave << 4) + l15; // output channel o for A fragment
  const float* Arow = A + row * 128;
  v8f accY[4] = {}; v8f accS[4] = {};
  for (int k0 = 0; k0 < 128; k0 += 4) {
    int kb = k0 + (hi << 1);
    v2f a;  a.x  = Arow[kb];           a.y  = Arow[kb + 1];
    v2f a2; a2.x = a.x * sS[kb];       a2.y = a.y * sS[kb + 1];
#pragma unroll
    for (int j = 0; j < 4; ++j) {
      v2f bf;
      bf.x = Xs[kb * XP + (j << 4) + l15];
      bf.y = Xs[(kb + 1) * XP + (j << 4) + l15];
      accY[j] = __builtin_amdgcn_wmma_f32_16x16x4_f32(
          false, a,  false, bf, (short)0, accY[j], false, false);
      accS[j] = __builtin_amdgcn_wmma_f32_16x16x4_f32(
          false, a2, false, bf, (short)0, accS[j], false, false);
    }
  }
  float w1 = wts[1];
#pragma unroll
  for (int j = 0; j < 4; ++j) {
    int p = p0 + (j << 4) + l15;
    float gv = g[(b << 14) + p] * w1;
#pragma unroll
    for (int r = 0; r < 8; ++r) {
      int o = (wave << 4) + (hi << 3) + r;   // C/D layout: M=r / M=r+8 halves
      out[((size_t)((b << 7) | o) << 14) + p] = accS[j][r] + gv * accY[j][r] + bS[o];
    }
  }
}

// ---------------------------------------------------------------------------
extern "C" void kernel_launch(void* const* d_in, const int* in_sizes, int n_in,
                              void* d_out, int out_size, void* d_ws, size_t ws_size,
                              hipStream_t stream) {
  (void)in_sizes; (void)n_in; (void)out_size; (void)ws_size;
  const float* x      = (const float*)d_in[0];
  const float* ca_w1  = (const float*)d_in[1];
  const float* ca_b1  = (const float*)d_in[2];
  const float* ca_w2  = (const float*)d_in[3];
  const float* ca_b2  = (const float*)d_in[4];
  const float* sa_w   = (const float*)d_in[5];
  const float* sa_b   = (const float*)d_in[6];
  const float* fa_w1  = (const float*)d_in[7];
  const float* fa_b1  = (const float*)d_in[8];
  const float* fa_w2  = (const float*)d_in[9];
  const float* fa_b2  = (const float*)d_in[10];
  const float* gmw    = (const float*)d_in[11];
  const float* gmb    = (const float*)d_in[12];
  const float* t1w    = (const float*)d_in[13];
  const float* t1b    = (const float*)d_in[14];
  const float* lng    = (const float*)d_in[15];
  const float* lnb    = (const float*)d_in[16];
  const float* t2w    = (const float*)d_in[17];
  const float* t2b    = (const float*)d_in[18];
  const float* alpha  = (const float*)d_in[19];
  const float* outw   = (const float*)d_in[20];
  const float* outb   = (const float*)d_in[21];
  float* out = (float*)d_out;

  float* ws    = (float*)d_ws;
  float* cw    = ws + CW_OFF;
  float* gap   = ws + GAP_OFF;
  float* fG    = ws + FG_OFF;
  float* ctx   = ws + CTX_OFF;
  float* sArr  = ws + S_OFF;
  float* bias2 = ws + BIAS2_OFF;
  float* wts   = ws + WTS_OFF;
  float* m     = ws + M_OFF;
  float* g     = ws + G_OFF;

  k_cw<<<1, 128, 0, stream>>>(cw);
  k_reduce<<<BB * CC, 256, 0, stream>>>(x, cw, gap, fG);
  k_mask<<<BB * (HW / 256), 256, 0, stream>>>(x, gmw, gmb, m);
  k_softmax<<<BB, 256, 0, stream>>>(m);
  k_ctx<<<BB * CC, 256, 0, stream>>>(x, m, ctx);
  k_mlp<<<BB, 128, 0, stream>>>(gap, fG, ctx,
                                ca_w1, ca_b1, ca_w2, ca_b2,
                                fa_w1, fa_b1, fa_w2, fa_b2,
                                t1w, t1b, lng, lnb, t2w, t2b,
                                alpha, outw, outb,
                                sArr, bias2, wts);
  k_conv<<<BB * 16, 256, 0, stream>>>(x, sa_w, sa_b, g);
  k_gemm<<<BB * 256, 256, 0, stream>>>(x, outw, sArr, bias2, wts, g, out);
}